// SelfAttention_34961033789783
// MI455X (gfx1250) — compile-verified
//
#include <hip/hip_runtime.h>
#include <hip/hip_bf16.h>

// ---------------------------------------------------------------------------
// Problem constants (from the reference)
// ---------------------------------------------------------------------------
#define NB        128     // batches
#define SEQ       64      // nodes per batch
#define NNODES    (NB * SEQ)        // 8192
#define IN_DIM    128
#define QKV_W     384     // N_HEADS * 3 * KEY_DIM
#define NHEADS    4
#define KDIM      32
#define VDIM      128     // NHEADS * KDIM

typedef __attribute__((ext_vector_type(2))) float v2f;
typedef __attribute__((ext_vector_type(8))) float v8f;

// D = A(16x4,f32) * B(4x16,f32) + C(16x16,f32)  -- wave32 CDNA5 WMMA
__device__ __forceinline__ v8f wmma4(v2f a, v2f b, v8f c) {
    return __builtin_amdgcn_wmma_f32_16x16x4_f32(
        /*neg_a=*/false, a, /*neg_b=*/false, b,
        /*c_mod=*/(short)0, c, /*reuse_a=*/false, /*reuse_b=*/false);
}

// ---------------------------------------------------------------------------
// fp32 WMMA GEMM + bias:  C[M,Ncol] = A[M,K] * B[K,Ncol] + bias[Ncol]
// Block = 128 threads (4 waves) owning a 64x16 macro-tile:
//   - B column strip (K x 16) staged once in LDS, transposed + padded
//   - wave w computes rows (block_m*64 + w*16 .. +15), the shared 16 cols
// Inner loop per WMMA: 1 global_load_b64 (imm offset) + 1 ds_load_b64 + wmma.
// K is a compile-time constant (128 here) -> fully unrolled.
// ---------------------------------------------------------------------------
template<int K>
__global__ void gemm_bias_wmma_f32(const float* __restrict__ A,
                                   const float* __restrict__ B,
                                   const float* __restrict__ bias,
                                   float* __restrict__ C,
                                   int M, int Ncol) {
    // Bs[n][k]: row stride 132 floats => 8B-aligned v2f rows, banks (4n+k)%64
    __shared__ float Bs[16][K + 4];

    const int lane   = threadIdx.x & 31;
    const int wave   = threadIdx.x >> 5;
    const int tilesN = Ncol >> 4;
    const int bm     = blockIdx.x / tilesN;       // 64-row macro tile index
    const int bn     = blockIdx.x - bm * tilesN;  // 16-col tile index

    // ---- stage B strip (K x 16) into LDS, transposed; coalesced 16-float rows
    for (int idx = threadIdx.x; idx < K * 16; idx += 128) {
        const int k = idx >> 4;
        const int n = idx & 15;
        Bs[n][k] = B[(size_t)k * Ncol + bn * 16 + n];
    }
    __syncthreads();

    const int l15 = lane & 15;
    const int kh  = (lane >> 4) << 1;             // 0 for lanes 0-15, 2 for 16-31
    const float* arow = A + (size_t)(bm * 64 + wave * 16 + l15) * K + kh;
    const float* brow = &Bs[l15][kh];

    v8f acc = {};
    #pragma unroll
    for (int k = 0; k < K; k += 4) {
        v2f a = *(const v2f*)(arow + k);          // A[m][k+kh .. +1] (imm offset)
        v2f b = *(const v2f*)(brow + k);          // B^T row in LDS   (imm offset)
        acc = wmma4(a, b, acc);
    }

    // ---- bias + store, pointer-stepped (no per-row multiply)
    const float bv = bias[bn * 16 + l15];
    float* crow = C + (size_t)(bm * 64 + wave * 16 + ((lane >> 4) << 3)) * Ncol
                    + bn * 16 + l15;
    #pragma unroll
    for (int r = 0; r < 8; ++r) {
        *crow = acc[r] + bv;
        crow += Ncol;
    }
}

// ---------------------------------------------------------------------------
// Attention: one 128-thread block per (batch, head).
//   P = softmax(Q K^T / sqrt(n_nodes[b])),  O = P V
// qkv layout: [node][384] with q at +0, k at +128, v at +256 (then h*32+d).
// attn output layout: [node][VDIM] with head h at columns h*32..h*32+31.
// ---------------------------------------------------------------------------
__global__ void attn_wmma_f32(const float* __restrict__ qkv,
                              const float* __restrict__ n_nodes,
                              float* __restrict__ attn) {
    __shared__ float Qs[SEQ][KDIM];
    __shared__ float Ks[SEQ][KDIM];
    __shared__ float Vs[SEQ][KDIM];
    __shared__ float Ps[SEQ][66];          // 66 = even pad -> 8B-aligned v2f rows

    const int bb   = blockIdx.x >> 2;      // batch
    const int h    = blockIdx.x & 3;       // head
    const int lane = threadIdx.x & 31;
    const int wave = threadIdx.x >> 5;

    // Stage Q, K, V tiles for this (batch, head) into LDS.
    for (int idx = threadIdx.x; idx < SEQ * KDIM; idx += 128) {
        const int r = idx >> 5, d = idx & 31;
        const float* p = qkv + (size_t)(bb * SEQ + r) * QKV_W + h * KDIM + d;
        Qs[r][d] = p[0];
        Ks[r][d] = p[IN_DIM];
        Vs[r][d] = p[2 * IN_DIM];
    }
    __syncthreads();

    const float scale = rsqrtf(n_nodes[bb]);   // = 1/8 for S = 64
    const int l15  = lane & 15;
    const int kh   = (lane >> 4) << 1;
    const int mrow = wave * 16 + l15;          // this wave owns rows wave*16..+15

    // ---- logits strip:  Ps[wave*16 .. +15][0..63] = (Q K^T) * scale ----
    #pragma unroll
    for (int nt = 0; nt < 4; ++nt) {
        v8f acc = {};
        const int ncol = nt * 16 + l15;
        #pragma unroll
        for (int k = 0; k < KDIM; k += 4) {
            v2f a = *(const v2f*)&Qs[mrow][k + kh];   // A = Q strip (row-major)
            v2f b = *(const v2f*)&Ks[ncol][k + kh];   // B = K^T -> contiguous rows
            acc = wmma4(a, b, acc);
        }
        #pragma unroll
        for (int r = 0; r < 8; ++r)
            Ps[wave * 16 + r + ((lane >> 4) << 3)][ncol] = acc[r] * scale;
    }
    __syncthreads();

    // ---- row softmax (threads 0..63 each own one row; wave-uniform branch) ----
    if (threadIdx.x < SEQ) {
        const int t = threadIdx.x;
        float mx = -3.402823466e38f;
        #pragma unroll 8
        for (int j = 0; j < SEQ; ++j) mx = fmaxf(mx, Ps[t][j]);
        float sum = 0.0f;
        #pragma unroll 8
        for (int j = 0; j < SEQ; ++j) {
            const float e = __expf(Ps[t][j] - mx);
            Ps[t][j] = e;
            sum += e;
        }
        const float inv = 1.0f / (sum + 1e-16f);
        #pragma unroll 8
        for (int j = 0; j < SEQ; ++j) Ps[t][j] *= inv;
    }
    __syncthreads();

    // ---- O strip = P(16x64) * V(64x32) ----
    #pragma unroll
    for (int ct = 0; ct < 2; ++ct) {
        v8f acc = {};
        const int ncol = ct * 16 + l15;
        #pragma unroll
        for (int k = 0; k < SEQ; k += 4) {
            v2f a = *(const v2f*)&Ps[mrow][k + kh];
            v2f b;
            b[0] = Vs[k + kh + 0][ncol];
            b[1] = Vs[k + kh + 1][ncol];
            acc = wmma4(a, b, acc);
        }
        #pragma unroll
        for (int r = 0; r < 8; ++r)
            attn[(size_t)(bb * SEQ + wave * 16 + r + ((lane >> 4) << 3)) * VDIM
                 + h * KDIM + ncol] = acc[r];
    }
}

// ---------------------------------------------------------------------------
// Launcher
// Inputs (setup_inputs order): x, W_qkv, b_qkv, W_out, b_out, n_nodes,
//                              grid_x, grid_y, batch2 (index arrays implicit).
// ---------------------------------------------------------------------------
extern "C" void kernel_launch(void* const* d_in, const int* in_sizes, int n_in,
                              void* d_out, int out_size, void* d_ws, size_t ws_size,
                              hipStream_t stream) {
    const float* x       = (const float*)d_in[0];
    const float* W_qkv   = (const float*)d_in[1];
    const float* b_qkv   = (const float*)d_in[2];
    const float* W_out   = (const float*)d_in[3];
    const float* b_out   = (const float*)d_in[4];
    const float* n_nodes = (const float*)d_in[5];

    float* qkv  = (float*)d_ws;                       // [8192 x 384]
    float* attn = qkv + (size_t)NNODES * QKV_W;       // [8192 x 128]

    // 1) qkv = x @ W_qkv + b_qkv : (8192/64) * (384/16) = 128*24 = 3072 blocks
    gemm_bias_wmma_f32<IN_DIM><<<dim3(3072), dim3(128), 0, stream>>>(
        x, W_qkv, b_qkv, qkv, NNODES, QKV_W);

    // 2) per-(batch, head) softmax attention: 128*4 = 512 blocks
    attn_wmma_f32<<<dim3(NB * NHEADS), dim3(128), 0, stream>>>(qkv, n_nodes, attn);

    // 3) out = attn @ W_out + b_out : (8192/64) * (128/16) = 128*8 = 1024 blocks
    gemm_bias_wmma_f32<VDIM><<<dim3(1024), dim3(128), 0, stream>>>(
        attn, W_out, b_out, (float*)d_out, NNODES, VDIM);
}